// LogEncoder_16527034155121
// MI455X (gfx1250) — compile-verified
//
#include <hip/hip_runtime.h>

typedef __attribute__((ext_vector_type(2))) float v2f;
typedef __attribute__((ext_vector_type(8))) float v8f;

#define SCALE_RANGE 16
#define BLOCK 256
#define F_DIM 64
#define O_DIM 128
#define LDS_STRIDE 17   // 16 + 1 pad to avoid LDS bank conflicts

__global__ __launch_bounds__(BLOCK)
void logenc_wmma_kernel(const float* __restrict__ x,
                        const float* __restrict__ W,     // [O=128, S=16] row-major
                        const float* __restrict__ bias,  // [O=128]
                        float* __restrict__ out) {       // [B, O=128]
    __shared__ float gsm[BLOCK * LDS_STRIDE];

    const int tid = threadIdx.x;
    const int row = blockIdx.x * BLOCK + tid;

    // ---------------- Phase 1: digit extraction + feature reduction ----------------
    // g[s] = sum_f mags[s, row, f]; recurrence: cur = x*0.1; mag=cur; cur=(cur-floor(cur))*10
    float g[SCALE_RANGE];
    #pragma unroll
    for (int s = 0; s < SCALE_RANGE; ++s) g[s] = 0.0f;

    const float4* xrow = reinterpret_cast<const float4*>(x + (size_t)row * F_DIM);
    for (int j = 0; j < F_DIM / 4; ++j) {
        float4 v = xrow[j];
        float e[4] = {v.x, v.y, v.z, v.w};
        #pragma unroll
        for (int k = 0; k < 4; ++k) {
            float cur = e[k] * 0.1f;
            #pragma unroll
            for (int s = 0; s < SCALE_RANGE; ++s) {
                g[s] += cur;
                cur = (cur - floorf(cur)) * 10.0f;
            }
        }
    }

    #pragma unroll
    for (int s = 0; s < SCALE_RANGE; ++s) gsm[tid * LDS_STRIDE + s] = g[s];
    __syncthreads();

    // ---------------- Phase 2: [B,16] x [16,128] GEMM via V_WMMA_F32_16X16X4_F32 ----------------
    const int wave  = tid >> 5;          // 8 waves per block, each owns 32 rows
    const int lane  = tid & 31;
    const int lhalf = lane >> 4;         // 0: lanes 0-15 (K even pair), 1: lanes 16-31
    const int lmod  = lane & 15;

    for (int rt = 0; rt < 2; ++rt) {     // two 16-row tiles per wave
        const int lrow     = wave * 32 + rt * 16 + lmod;            // LDS-local row (M = lmod)
        const int gRowBase = blockIdx.x * BLOCK + wave * 32 + rt * 16;

        // A-matrix 16x4 f32 layout: VGPR0 = K {0 | 2}, VGPR1 = K {1 | 3} per half-wave.
        v2f a[4];
        #pragma unroll
        for (int ks = 0; ks < 4; ++ks) {
            const int kbase = ks * 4 + lhalf * 2;
            a[ks].x = gsm[lrow * LDS_STRIDE + kbase];
            a[ks].y = gsm[lrow * LDS_STRIDE + kbase + 1];
        }

        for (int nt = 0; nt < 8; ++nt) { // 8 N-tiles cover O=128
            const int col = nt * 16 + lmod;

            // All 8 C components of a lane map to the same column -> broadcast 64*bias[col].
            const float bf = 64.0f * bias[col];
            v8f c;
            #pragma unroll
            for (int i = 0; i < 8; ++i) c[i] = bf;

            // B = W^T slice: B[k][n] = W[col][k], mirrored half-wave K layout.
            #pragma unroll
            for (int ks = 0; ks < 4; ++ks) {
                const int kbase = ks * 4 + lhalf * 2;
                v2f bv;
                bv.x = W[col * SCALE_RANGE + kbase];
                bv.y = W[col * SCALE_RANGE + kbase + 1];
                // 8 args: (neg_a, A, neg_b, B, c_mod, C, reuse_a, reuse_b)
                c = __builtin_amdgcn_wmma_f32_16x16x4_f32(false, a[ks], false, bv,
                                                          (short)0, c, false, false);
            }

            // C/D layout: VGPR i, lane l -> row = i + 8*(l>=16), col = l%16.
            #pragma unroll
            for (int i = 0; i < 8; ++i) {
                const int r = gRowBase + i + 8 * lhalf;
                out[(size_t)r * O_DIM + col] = c[i];
            }
        }
    }
}

extern "C" void kernel_launch(void* const* d_in, const int* in_sizes, int n_in,
                              void* d_out, int out_size, void* d_ws, size_t ws_size,
                              hipStream_t stream) {
    const float* x    = (const float*)d_in[0];  // [B, 64]
    const float* W    = (const float*)d_in[1];  // [128, 16]
    const float* bias = (const float*)d_in[2];  // [128]
    float* out        = (float*)d_out;          // [B, 128]

    const int B = in_sizes[0] / F_DIM;          // 65536
    dim3 grid(B / BLOCK);                       // 256 blocks x 256 threads, 1 row/thread
    logenc_wmma_kernel<<<grid, BLOCK, 0, stream>>>(x, W, bias, out);
}